// LateTokenSheafReadoutTransport_31842887533234
// MI455X (gfx1250) — compile-verified
//
#include <hip/hip_runtime.h>
#include <math.h>

typedef __attribute__((ext_vector_type(2))) float v2f;
typedef __attribute__((ext_vector_type(8))) float v8f;

#define EPS_WHITEN 1e-4f
#define LN_EPS     1e-5f

__device__ __forceinline__ float gelu_exact(float x) {
  return 0.5f * x * (1.0f + erff(x * 0.70710678118654752f));
}

__device__ __forceinline__ float wave_sum32(float v) {
#pragma unroll
  for (int m = 1; m < 32; m <<= 1) v += __shfl_xor(v, m, 32);
  return v;
}

__device__ __forceinline__ float det4(const float m[4][4]) {
  float s0 = m[0][0]*m[1][1] - m[0][1]*m[1][0];
  float s1 = m[0][0]*m[1][2] - m[0][2]*m[1][0];
  float s2 = m[0][0]*m[1][3] - m[0][3]*m[1][0];
  float s3 = m[0][1]*m[1][2] - m[0][2]*m[1][1];
  float s4 = m[0][1]*m[1][3] - m[0][3]*m[1][1];
  float s5 = m[0][2]*m[1][3] - m[0][3]*m[1][2];
  float c5 = m[2][2]*m[3][3] - m[2][3]*m[3][2];
  float c4 = m[2][1]*m[3][3] - m[2][3]*m[3][1];
  float c3 = m[2][1]*m[3][2] - m[2][2]*m[3][1];
  float c2 = m[2][0]*m[3][3] - m[2][3]*m[3][0];
  float c1 = m[2][0]*m[3][2] - m[2][2]*m[3][0];
  float c0 = m[2][0]*m[3][1] - m[2][1]*m[3][0];
  return s0*c5 - s1*c4 + s2*c3 + s3*c2 - s4*c1 + s5*c0;
}

// ---------------------------------------------------------------------------
// Kernel A: wave per node. Whiten (Newton-Schulz S^-1/2) + Procrustes align
// (Jacobi SVD of 4x4 with proper-rotation det fix). H read exactly once,
// coalesced; Z_aligned written once.
// ---------------------------------------------------------------------------
__global__ void __launch_bounds__(256)
whiten_align_kernel(const float* __restrict__ H, const float* __restrict__ anchor,
                    float* __restrict__ Zal, int N) {
  const int wave = blockIdx.x * (blockDim.x >> 5) + (threadIdx.x >> 5);
  const int lane = threadIdx.x & 31;
  if (wave >= N) return;
  const float* __restrict__ hrow = H + (size_t)wave * 256;

  float hv[4][2], anc[4][2];
#pragma unroll
  for (int d = 0; d < 4; ++d) {
    hv[d][0]  = hrow[d * 64 + lane];
    hv[d][1]  = hrow[d * 64 + lane + 32];
    anc[d][0] = anchor[d * 64 + lane];
    anc[d][1] = anchor[d * 64 + lane + 32];
  }

  // S = H H^T + eps I  (wave-reduced over the 64 columns)
  float S[4][4];
#pragma unroll
  for (int d = 0; d < 4; ++d) {
#pragma unroll
    for (int e = 0; e <= d; ++e) {
      float p = hv[d][0] * hv[e][0] + hv[d][1] * hv[e][1];
      p = wave_sum32(p);
      if (d == e) p += EPS_WHITEN;
      S[d][e] = p; S[e][d] = p;
    }
  }

  // Coupled Newton-Schulz: A = S/tr(S); Y->A^(1/2), Zm->A^(-1/2)
  float tr = S[0][0] + S[1][1] + S[2][2] + S[3][3];
  float itr = 1.0f / tr;
  float Y[4][4], Zm[4][4];
#pragma unroll
  for (int i = 0; i < 4; ++i)
#pragma unroll
    for (int j = 0; j < 4; ++j) {
      Y[i][j]  = S[i][j] * itr;
      Zm[i][j] = (i == j) ? 1.0f : 0.0f;
    }
#pragma unroll
  for (int it = 0; it < 14; ++it) {
    float T[4][4];
#pragma unroll
    for (int i = 0; i < 4; ++i)
#pragma unroll
      for (int j = 0; j < 4; ++j) {
        float a = (i == j) ? 3.0f : 0.0f;
#pragma unroll
        for (int k = 0; k < 4; ++k) a -= Zm[i][k] * Y[k][j];
        T[i][j] = a;
      }
    float Yn[4][4], Zn[4][4];
#pragma unroll
    for (int i = 0; i < 4; ++i)
#pragma unroll
      for (int j = 0; j < 4; ++j) {
        float ay = 0.0f, az = 0.0f;
#pragma unroll
        for (int k = 0; k < 4; ++k) { ay += Y[i][k] * T[k][j]; az += T[i][k] * Zm[k][j]; }
        Yn[i][j] = 0.5f * ay; Zn[i][j] = 0.5f * az;
      }
#pragma unroll
    for (int i = 0; i < 4; ++i)
#pragma unroll
      for (int j = 0; j < 4; ++j) { Y[i][j] = Yn[i][j]; Zm[i][j] = Zn[i][j]; }
  }
  float isqt = 1.0f / sqrtf(tr);
  float W[4][4];
#pragma unroll
  for (int i = 0; i < 4; ++i)
#pragma unroll
    for (int j = 0; j < 4; ++j) W[i][j] = Zm[i][j] * isqt;  // S^(-1/2)

  // z = W h (per column); M[d][e] = sum_c anchor[d][c] * z[e][c]
  float z[4][2];
#pragma unroll
  for (int d = 0; d < 4; ++d) {
    float a0 = 0.0f, a1 = 0.0f;
#pragma unroll
    for (int e = 0; e < 4; ++e) { a0 += W[d][e] * hv[e][0]; a1 += W[d][e] * hv[e][1]; }
    z[d][0] = a0; z[d][1] = a1;
  }
  float M[4][4];
#pragma unroll
  for (int d = 0; d < 4; ++d)
#pragma unroll
    for (int e = 0; e < 4; ++e) {
      float p = anc[d][0] * z[e][0] + anc[d][1] * z[e][1];
      M[d][e] = wave_sum32(p);
    }

  // Jacobi eigendecomposition of A2 = M^T M -> V, sigma^2
  float A2[4][4], V[4][4];
#pragma unroll
  for (int i = 0; i < 4; ++i)
#pragma unroll
    for (int j = 0; j < 4; ++j) {
      float a = 0.0f;
#pragma unroll
      for (int k = 0; k < 4; ++k) a += M[k][i] * M[k][j];
      A2[i][j] = a;
      V[i][j] = (i == j) ? 1.0f : 0.0f;
    }
  const int JP[6] = {0, 0, 0, 1, 1, 2};
  const int JQ[6] = {1, 2, 3, 2, 3, 3};
#pragma unroll
  for (int sweep = 0; sweep < 6; ++sweep) {
#pragma unroll
    for (int r = 0; r < 6; ++r) {
      const int p = JP[r], q = JQ[r];
      float apq = A2[p][q];
      float app = A2[p][p], aqq = A2[q][q];
      if (fabsf(apq) > 1e-24f * (fabsf(app) + fabsf(aqq) + 1e-30f)) {
        float theta = (aqq - app) / (2.0f * apq);
        float t = copysignf(1.0f, theta) / (fabsf(theta) + sqrtf(theta * theta + 1.0f));
        float c = 1.0f / sqrtf(t * t + 1.0f);
        float s = t * c;
#pragma unroll
        for (int k = 0; k < 4; ++k) {
          float akp = A2[k][p], akq = A2[k][q];
          A2[k][p] = c * akp - s * akq;
          A2[k][q] = s * akp + c * akq;
        }
#pragma unroll
        for (int k = 0; k < 4; ++k) {
          float apk = A2[p][k], aqk = A2[q][k];
          A2[p][k] = c * apk - s * aqk;
          A2[q][k] = s * apk + c * aqk;
        }
#pragma unroll
        for (int k = 0; k < 4; ++k) {
          float vkp = V[k][p], vkq = V[k][q];
          V[k][p] = c * vkp - s * vkq;
          V[k][q] = s * vkp + c * vkq;
        }
      }
    }
  }

  float sig[4], isig[4];
#pragma unroll
  for (int i = 0; i < 4; ++i) {
    sig[i]  = sqrtf(fmaxf(A2[i][i], 0.0f));
    isig[i] = 1.0f / fmaxf(sig[i], 1e-20f);
  }
  int jmin = 0; float smin = sig[0];
#pragma unroll
  for (int i = 1; i < 4; ++i) { if (sig[i] < smin) { smin = sig[i]; jmin = i; } }
  float flip = (det4(M) < 0.0f) ? -1.0f : 1.0f;

  // U = M V Sigma^-1, flip min-sigma column; R = U V^T; G = R W
  float U[4][4];
#pragma unroll
  for (int k = 0; k < 4; ++k)
#pragma unroll
    for (int i = 0; i < 4; ++i) {
      float a = 0.0f;
#pragma unroll
      for (int j = 0; j < 4; ++j) a += M[k][j] * V[j][i];
      U[k][i] = a * isig[i] * ((i == jmin) ? flip : 1.0f);
    }
  float R[4][4];
#pragma unroll
  for (int i = 0; i < 4; ++i)
#pragma unroll
    for (int j = 0; j < 4; ++j) {
      float a = 0.0f;
#pragma unroll
      for (int k = 0; k < 4; ++k) a += U[i][k] * V[j][k];
      R[i][j] = a;
    }
  float G[4][4];
#pragma unroll
  for (int i = 0; i < 4; ++i)
#pragma unroll
    for (int j = 0; j < 4; ++j) {
      float a = 0.0f;
#pragma unroll
      for (int k = 0; k < 4; ++k) a += R[i][k] * W[k][j];
      G[i][j] = a;
    }

  float* __restrict__ zo = Zal + (size_t)wave * 256;
#pragma unroll
  for (int d = 0; d < 4; ++d) {
    float a0 = 0.0f, a1 = 0.0f;
#pragma unroll
    for (int e = 0; e < 4; ++e) { a0 += G[d][e] * hv[e][0]; a1 += G[d][e] * hv[e][1]; }
    zo[d * 64 + lane]      = a0;
    zo[d * 64 + lane + 32] = a1;
  }
}

// ---------------------------------------------------------------------------
// Kernel B: wave per 16 nodes. u = sum_d Zal^2 (rotation-invariant, so Zal is
// fine) built directly in the V_WMMA_F32_16X16X4_F32 A-fragment layout
// (lanes 0-15: K=0,1; lanes 16-31: K=2,3). 16 K-chunks x 4 N-tiles of f32
// WMMA, then exact gelu, w2 dot, half-wave shuffle reduce -> per-node score.
// ---------------------------------------------------------------------------
__global__ void __launch_bounds__(256)
node_mlp_scores_kernel(const float* __restrict__ Zal,
                       const float* __restrict__ w1, const float* __restrict__ b1,
                       const float* __restrict__ w2, const float* __restrict__ b2,
                       float* __restrict__ scores, int N) {
  const int wave  = blockIdx.x * (blockDim.x >> 5) + (threadIdx.x >> 5);
  const int lane  = threadIdx.x & 31;
  const int half  = lane >> 4;
  const int lr    = lane & 15;
  const int node0 = wave * 16;
  if (node0 >= N) return;

  // u values this lane contributes to the A fragments: channels 4k+2*half{,+1}
  const float* __restrict__ zrow = Zal + (size_t)(node0 + lr) * 256;
  float ua[16][2];
#pragma unroll
  for (int k = 0; k < 16; ++k) {
    const int c = 4 * k + 2 * half;
    float u0 = 0.0f, u1 = 0.0f;
#pragma unroll
    for (int d = 0; d < 4; ++d) {
      float z0 = zrow[d * 64 + c];
      float z1 = zrow[d * 64 + c + 1];
      u0 += z0 * z0; u1 += z1 * z1;
    }
    ua[k][0] = u0; ua[k][1] = u1;
  }

  v8f acc[4];
#pragma unroll
  for (int t = 0; t < 4; ++t)
#pragma unroll
    for (int v = 0; v < 8; ++v) acc[t][v] = 0.0f;

#pragma unroll
  for (int k = 0; k < 16; ++k) {
    v2f a; a[0] = ua[k][0]; a[1] = ua[k][1];
    const int r0 = 4 * k + 2 * half;   // B rows this lane supplies
#pragma unroll
    for (int t = 0; t < 4; ++t) {
      v2f bf;
      bf[0] = w1[(r0    ) * 64 + 16 * t + lr];
      bf[1] = w1[(r0 + 1) * 64 + 16 * t + lr];
      acc[t] = __builtin_amdgcn_wmma_f32_16x16x4_f32(
          false, a, false, bf, (short)0, acc[t], false, false);
    }
  }

  // Epilogue: gelu(h) dot w2, reduce over the 16 lanes of each half-wave.
  float p[8];
#pragma unroll
  for (int v = 0; v < 8; ++v) p[v] = 0.0f;
#pragma unroll
  for (int t = 0; t < 4; ++t) {
    const float bb = b1[16 * t + lr];
    const float ww = w2[16 * t + lr];
#pragma unroll
    for (int v = 0; v < 8; ++v) {
      float x = acc[t][v] + bb;
      p[v] += gelu_exact(x) * ww;
    }
  }
#pragma unroll
  for (int v = 0; v < 8; ++v) {
#pragma unroll
    for (int m = 1; m < 16; m <<= 1) p[v] += __shfl_xor(p[v], m, 32);
  }
  if (lr == 0) {
    const float bb2 = b2[0];
#pragma unroll
    for (int v = 0; v < 8; ++v)
      scores[node0 + v + 8 * half] = p[v] + bb2;  // D layout: M = v + 8*(lane>=16)
  }
}

// ---------------------------------------------------------------------------
// Kernel C0: segment starts by binary search over sorted batch ids.
// ---------------------------------------------------------------------------
__global__ void seg_starts_kernel(const int* __restrict__ batch, int* __restrict__ start,
                                  int N, int B) {
  int b = blockIdx.x * blockDim.x + threadIdx.x;
  if (b > B) return;
  if (b == B) { start[B] = N; return; }
  int lo = 0, hi = N;
  while (lo < hi) {
    int mid = (lo + hi) >> 1;
    if (batch[mid] < b) lo = mid + 1; else hi = mid;
  }
  start[b] = lo;
}

// ---------------------------------------------------------------------------
// Kernel C1: per-graph softmax stats (max, sum of exp).
// ---------------------------------------------------------------------------
__global__ void __launch_bounds__(256)
seg_softmax_stats_kernel(const float* __restrict__ scores, const int* __restrict__ start,
                         float* __restrict__ smax, float* __restrict__ dsum) {
  const int b = blockIdx.x;
  const int s = start[b], e = start[b + 1];
  __shared__ float red[256];
  float m = -3.4e38f;
  for (int i = s + threadIdx.x; i < e; i += 256) m = fmaxf(m, scores[i]);
  red[threadIdx.x] = m; __syncthreads();
  for (int off = 128; off > 0; off >>= 1) {
    if (threadIdx.x < off) red[threadIdx.x] = fmaxf(red[threadIdx.x], red[threadIdx.x + off]);
    __syncthreads();
  }
  m = red[0]; __syncthreads();
  float sum = 0.0f;
  for (int i = s + threadIdx.x; i < e; i += 256) sum += expf(scores[i] - m);
  red[threadIdx.x] = sum; __syncthreads();
  for (int off = 128; off > 0; off >>= 1) {
    if (threadIdx.x < off) red[threadIdx.x] += red[threadIdx.x + off];
    __syncthreads();
  }
  if (threadIdx.x == 0) { smax[b] = m; dsum[b] = red[0]; }
}

// ---------------------------------------------------------------------------
// Kernel D: per-graph weighted pooling; softmax weights staged in LDS,
// coalesced Z_aligned reads (thread t owns element (d,c)=t of the 4x64 tile).
// ---------------------------------------------------------------------------
__global__ void __launch_bounds__(256)
pool_kernel(const float* __restrict__ Zal, const float* __restrict__ scores,
            const int* __restrict__ start, const float* __restrict__ smax,
            const float* __restrict__ dsum, float* __restrict__ pooled) {
  const int b = blockIdx.x;
  const int t = threadIdx.x;
  const int s = start[b], e = start[b + 1];
  const float m = smax[b];
  const float dd = dsum[b];
  const float inv = (dd > 0.0f) ? (1.0f / dd) : 0.0f;
  __shared__ float wsm[256];
  float acc = 0.0f;
  for (int base = s; base < e; base += 256) {
    const int n = base + t;
    wsm[t] = (n < e) ? expf(scores[n] - m) * inv : 0.0f;
    __syncthreads();
    const int cnt = min(256, e - base);
    for (int j = 0; j < cnt; ++j)
      acc += Zal[(size_t)(base + j) * 256 + t] * wsm[j];
    __syncthreads();
  }
  pooled[(size_t)b * 256 + t] = acc;
}

// ---------------------------------------------------------------------------
// Kernel E: per-graph head: concat(token,pooled)@wc1 -> gelu -> @wc2 -> LN ->
// sum_d square. 64 threads (one per channel), hc/y staged in LDS.
// ---------------------------------------------------------------------------
__global__ void __launch_bounds__(64)
graph_head_kernel(const float* __restrict__ pooled, const float* __restrict__ token,
                  const float* __restrict__ wc1, const float* __restrict__ bc1,
                  const float* __restrict__ wc2, const float* __restrict__ bc2,
                  const float* __restrict__ ln_g, const float* __restrict__ ln_b,
                  float* __restrict__ out) {
  const int b = blockIdx.x;
  const int j = threadIdx.x;  // 0..63
  __shared__ float hc[4][64];
  __shared__ float ys[4][64];

#pragma unroll
  for (int d = 0; d < 4; ++d) {
    float acc = bc1[j];
    for (int k = 0; k < 64; ++k)
      acc += token[d * 64 + k] * wc1[k * 64 + j];
    for (int k = 0; k < 64; ++k)
      acc += pooled[(size_t)b * 256 + d * 64 + k] * wc1[(64 + k) * 64 + j];
    hc[d][j] = gelu_exact(acc);
  }
  __syncthreads();

  float y[4];
#pragma unroll
  for (int d = 0; d < 4; ++d) {
    float a = bc2[j];
    for (int k = 0; k < 64; ++k) a += hc[d][k] * wc2[k * 64 + j];
    y[d] = a;
    ys[d][j] = a;
  }
  __syncthreads();

  float o = 0.0f;
#pragma unroll
  for (int d = 0; d < 4; ++d) {
    float s1 = 0.0f, s2 = 0.0f;
    for (int k = 0; k < 64; ++k) { float v = ys[d][k]; s1 += v; s2 += v * v; }
    float mu  = s1 * (1.0f / 64.0f);
    float var = s2 * (1.0f / 64.0f) - mu * mu;
    float u = (y[d] - mu) * rsqrtf(var + LN_EPS) * ln_g[j] + ln_b[j];
    o += u * u;
  }
  out[(size_t)b * 64 + j] = o;
}

// ---------------------------------------------------------------------------
extern "C" void kernel_launch(void* const* d_in, const int* in_sizes, int n_in,
                              void* d_out, int out_size, void* d_ws, size_t ws_size,
                              hipStream_t stream) {
  const float* H      = (const float*)d_in[0];
  const int*   batch  = (const int*)d_in[1];
  // d_in[2] = num_graphs (device scalar; B recovered host-side from out_size)
  const float* anchor = (const float*)d_in[3];
  const float* w1     = (const float*)d_in[4];
  const float* b1     = (const float*)d_in[5];
  const float* w2     = (const float*)d_in[6];
  const float* b2     = (const float*)d_in[7];
  const float* token  = (const float*)d_in[8];
  const float* wc1    = (const float*)d_in[9];
  const float* bc1    = (const float*)d_in[10];
  const float* wc2    = (const float*)d_in[11];
  const float* bc2    = (const float*)d_in[12];
  const float* ln_g   = (const float*)d_in[13];
  const float* ln_b   = (const float*)d_in[14];
  float* out = (float*)d_out;

  const int N = in_sizes[0] / 256;  // [N,4,64]
  const int B = out_size / 64;      // [B,64]

  // Workspace layout
  float* Zal    = (float*)d_ws;                 // N*256
  float* scores = Zal + (size_t)N * 256;        // N
  int*   start  = (int*)(scores + N);           // B+1
  float* smax   = (float*)(start + (B + 1));    // B
  float* dsum   = smax + B;                     // B
  float* pooled = dsum + B;                     // B*256

  const int wavesPerBlockA = 256 / 32;          // 8 nodes per block
  whiten_align_kernel<<<(N + wavesPerBlockA - 1) / wavesPerBlockA, 256, 0, stream>>>(
      H, anchor, Zal, N);

  const int nodesPerBlockB = (256 / 32) * 16;   // 128 nodes per block
  node_mlp_scores_kernel<<<(N + nodesPerBlockB - 1) / nodesPerBlockB, 256, 0, stream>>>(
      Zal, w1, b1, w2, b2, scores, N);

  seg_starts_kernel<<<(B + 1 + 255) / 256, 256, 0, stream>>>(batch, start, N, B);
  seg_softmax_stats_kernel<<<B, 256, 0, stream>>>(scores, start, smax, dsum);
  pool_kernel<<<B, 256, 0, stream>>>(Zal, scores, start, smax, dsum, pooled);
  graph_head_kernel<<<B, 64, 0, stream>>>(pooled, token, wc1, bc1, wc2, bc2,
                                          ln_g, ln_b, out);
}